// GumbelCodebook_7017976562322
// MI455X (gfx1250) — compile-verified
//
#include <hip/hip_runtime.h>
#include <stdint.h>

// GumbelCodebook forward: y = hard one-hot (with straight-through fp value
// (1-s)+s at the argmax, exact 0 elsewhere), z = yv * codebook[argmax].
// Pure streaming problem: ~800 MiB traffic -> ~34us floor at 23.3 TB/s.
// No WMMA: the one-hot einsum is a gather, matrix units cannot help.
// CDNA5 paths used: async global->LDS (ASYNCcnt), wave32 shuffles, NT stores.

#define NUM_CODES 2048
#define CODE_DIM  256
#define THREADS   256
#define NWAVES    (THREADS / 32)
#define VPT       (NUM_CODES / THREADS)   // 8 floats per thread
#define CHUNKS    (VPT / 4)               // 2 x 16B per input per thread

typedef float v4f __attribute__((ext_vector_type(4)));

static_assert(CODE_DIM == THREADS, "one lane per z element");

__global__ void __launch_bounds__(THREADS)
gumbel_st_kernel(const float* __restrict__ logits,
                 const float* __restrict__ gumbel,
                 const float* __restrict__ codebook,
                 float* __restrict__ z,     // [rows, CODE_DIM]
                 float* __restrict__ y)     // [rows, NUM_CODES]
{
    __shared__ __align__(16) float sl[NUM_CODES];
    __shared__ __align__(16) float sg[NUM_CODES];
    __shared__ float red_v[NWAVES];
    __shared__ int   red_i[NWAVES];
    __shared__ float red_s[NWAVES];

    const int    tid  = threadIdx.x;
    const size_t base = (size_t)blockIdx.x * NUM_CODES;

    // ---- Stage both 8KB input rows into LDS via CDNA5 async DMA ----
    {
        const uint32_t ldsl = (uint32_t)(uintptr_t)sl;
        const uint32_t ldsg = (uint32_t)(uintptr_t)sg;
#pragma unroll
        for (int i = 0; i < CHUNKS; ++i) {
            const uint32_t off = (uint32_t)((i * THREADS + tid) * 16);
            const uint64_t gl  = (uint64_t)(uintptr_t)logits + base * 4ull + off;
            const uint64_t gg  = (uint64_t)(uintptr_t)gumbel + base * 4ull + off;
            const uint32_t ll  = ldsl + off;
            const uint32_t lg  = ldsg + off;
            asm volatile("global_load_async_to_lds_b128 %0, %1, off"
                         :: "v"(ll), "v"(gl) : "memory");
            asm volatile("global_load_async_to_lds_b128 %0, %1, off"
                         :: "v"(lg), "v"(gg) : "memory");
        }
        asm volatile("s_wait_asynccnt 0" ::: "memory");
    }
    __syncthreads();

    // ---- Pass 1: x = logits + gumbel (tau=1); row max + first-argmax ----
    float x[VPT];
    float lmax = -__builtin_inff();
    int   lidx = 0;
#pragma unroll
    for (int i = 0; i < CHUNKS; ++i) {
        const v4f a = ((const v4f*)sl)[i * THREADS + tid];
        const v4f b = ((const v4f*)sg)[i * THREADS + tid];
#pragma unroll
        for (int j = 0; j < 4; ++j) {
            const float v = a[j] + b[j];
            x[i * 4 + j] = v;
            const int e = (i * THREADS + tid) * 4 + j;   // ascending per thread
            if (v > lmax) { lmax = v; lidx = e; }        // strict > keeps first
        }
    }
    // wave32 reduction, tie-break to smaller index (matches jnp.argmax)
#pragma unroll
    for (int o = 16; o > 0; o >>= 1) {
        const float ov = __shfl_xor(lmax, o, 32);
        const int   oi = __shfl_xor(lidx, o, 32);
        if (ov > lmax || (ov == lmax && oi < lidx)) { lmax = ov; lidx = oi; }
    }
    if ((tid & 31) == 0) { red_v[tid >> 5] = lmax; red_i[tid >> 5] = lidx; }
    __syncthreads();
    float rmax = red_v[0];
    int   ridx = red_i[0];
#pragma unroll
    for (int w = 1; w < NWAVES; ++w) {
        const float ov = red_v[w]; const int oi = red_i[w];
        if (ov > rmax || (ov == rmax && oi < ridx)) { rmax = ov; ridx = oi; }
    }

    // ---- Pass 2: sum of exp(x - max) (x still live in VGPRs) ----
    float lsum = 0.0f;
#pragma unroll
    for (int k = 0; k < VPT; ++k) lsum += expf(x[k] - rmax);
#pragma unroll
    for (int o = 16; o > 0; o >>= 1) lsum += __shfl_xor(lsum, o, 32);
    if ((tid & 31) == 0) red_s[tid >> 5] = lsum;
    __syncthreads();
    float tot = 0.0f;
#pragma unroll
    for (int w = 0; w < NWAVES; ++w) tot += red_s[w];   // same order in all lanes

    const float smax = 1.0f / tot;            // y_soft at the argmax (exp(0)=1)
    const float yv   = (1.0f - smax) + smax;  // straight-through forward value

    // ---- y: exact zeros everywhere except yv at ridx; NT streaming stores ----
    {
        v4f* yrow = (v4f*)(y + base);
#pragma unroll
        for (int i = 0; i < CHUNKS; ++i) {
            const int r = ridx - (i * THREADS + tid) * 4;
            v4f v;
            v[0] = (r == 0) ? yv : 0.0f;
            v[1] = (r == 1) ? yv : 0.0f;
            v[2] = (r == 2) ? yv : 0.0f;
            v[3] = (r == 3) ? yv : 0.0f;
            __builtin_nontemporal_store(v, yrow + i * THREADS + tid);
        }
    }

    // ---- z: gather codebook row (stays hot in the 192MB L2) and scale ----
    {
        const float c = codebook[(size_t)ridx * CODE_DIM + tid];
        __builtin_nontemporal_store(yv * c, z + (size_t)blockIdx.x * CODE_DIM + tid);
    }
}

extern "C" void kernel_launch(void* const* d_in, const int* in_sizes, int n_in,
                              void* d_out, int out_size, void* d_ws, size_t ws_size,
                              hipStream_t stream) {
    const float* logits   = (const float*)d_in[0];
    const float* gumbel   = (const float*)d_in[1];
    const float* codebook = (const float*)d_in[2];
    const int rows = in_sizes[0] / NUM_CODES;           // 8*4096 = 32768
    float* z = (float*)d_out;                           // [rows, 256] first
    float* y = z + (size_t)rows * CODE_DIM;             // then [rows, 2048]
    gumbel_st_kernel<<<dim3(rows), dim3(THREADS), 0, stream>>>(
        logits, gumbel, codebook, z, y);
    (void)n_in; (void)out_size; (void)d_ws; (void)ws_size;
}